// FADE_17910013624966
// MI455X (gfx1250) — compile-verified
//
#include <hip/hip_runtime.h>
#include <math.h>

#define BB 2
#define CC 256
#define HH 80
#define WW 80
#define EE 64
#define H2 160
#define W2 160
#define NPIX_HI (H2*W2)   // 25600 per image
#define NPIX_LO (HH*WW)   // 6400  per image

typedef __attribute__((ext_vector_type(2))) float v2f;
typedef __attribute__((ext_vector_type(8))) float v8f;
typedef unsigned int u32x4 __attribute__((ext_vector_type(4)));
typedef int          i32x4 __attribute__((ext_vector_type(4)));
typedef int          i32x8 __attribute__((ext_vector_type(8)));

// ---------------------------------------------------------------------------
// Kernel 1: 1x1 conv as GEMM via V_WMMA_F32_16X16X4_F32 (full f32 precision).
// Y[b,e,pix] = sum_c Wmat[e,c] * X[b,c,pix] (+ bias[e])
// One wave computes one 16(E) x 16(pixel) tile; 64 WMMA steps over K=256.
// NPIX and HAS_BIAS are compile-time: immediate-offset addressing, no
// per-element bias branches.
// ---------------------------------------------------------------------------
template <int NPIX, bool HAS_BIAS>
__global__ __launch_bounds__(128) void conv1x1_wmma(
    const float* __restrict__ Wmat, const float* __restrict__ X,
    float* __restrict__ Y, const float* __restrict__ bias)
{
  constexpr int ntPerImg = NPIX / 16;
  const int wave  = threadIdx.x >> 5;
  const int lane  = threadIdx.x & 31;
  const int tile  = blockIdx.x * 4 + wave;
  const int mtile = tile & 3;
  const int rest  = tile >> 2;
  const int b     = rest / ntPerImg;
  const int n0    = (rest % ntPerImg) * 16;
  const int m     = lane & 15;
  const int half  = lane >> 4;

  const float* Xb   = X + (size_t)b * CC * NPIX + n0 + m;
  const float* Arow = Wmat + (size_t)(mtile * 16 + m) * CC;

  v8f acc = {};
  #pragma unroll 8
  for (int kk = 0; kk < CC; kk += 4) {
    const int kA = kk + 2 * half;
    v2f a = *(const v2f*)(Arow + kA);         // contiguous, 8B aligned
    v2f bv;
    bv.x = Xb[(size_t)kA * NPIX];
    bv.y = Xb[(size_t)(kA + 1) * NPIX];
    acc = __builtin_amdgcn_wmma_f32_16x16x4_f32(
        /*neg_a=*/false, a, /*neg_b=*/false, bv,
        /*c_mod=*/(short)0, acc, /*reuse_a=*/false, /*reuse_b=*/false);
  }

  float* Yb = Y + (size_t)b * EE * NPIX + n0 + m;
  #pragma unroll
  for (int r = 0; r < 8; ++r) {
    const int e = mtile * 16 + r + 8 * half;
    float v = acc[r];
    if constexpr (HAS_BIAS) v += bias[e];
    Yb[(size_t)e * NPIX] = v;                 // coalesced over lanes 0..15
  }
}

// ---------------------------------------------------------------------------
// Kernel 2: gate = sigmoid(1x1 conv(de, gate_w) + gate_b), one value per
// low-res pixel (nearest-2x upsample makes it constant over each 2x2 block).
// ---------------------------------------------------------------------------
__global__ __launch_bounds__(256) void gate_kernel(
    const float* __restrict__ de, const float* __restrict__ gw,
    const float* __restrict__ gb, float* __restrict__ gate)
{
  const int tid = blockIdx.x * blockDim.x + threadIdx.x;
  if (tid >= BB * NPIX_LO) return;
  const int b   = tid / NPIX_LO;
  const int pix = tid % NPIX_LO;
  const float* dp = de + (size_t)b * CC * NPIX_LO + pix;
  float g = gb[0];
  for (int c = 0; c < CC; ++c) g += dp[(size_t)c * NPIX_LO] * gw[c];
  gate[tid] = 1.0f / (1.0f + __expf(-g));
}

// ---------------------------------------------------------------------------
// Kernel 3: semi-shift 3x3 convs + combine + softmax over 25.
// conv2_k (25x64x3x3 = 57.6 KB, contiguous) is DMA'd into LDS by the Tensor
// Data Mover (one wave issues a 1-row D# descriptor; completion via
// s_wait_tensorcnt), falling back to a plain cooperative copy where the
// builtin is unavailable. All weight reads in the 25-wide FMA loops are LDS
// broadcasts (same address across lanes), so a linear layout is optimal.
// ---------------------------------------------------------------------------
__global__ __launch_bounds__(256) void semishift_kernel(
    const float* __restrict__ ce, const float* __restrict__ cd,
    const float* __restrict__ k2w, const float* __restrict__ k2b,
    float* __restrict__ kern)
{
  __shared__ float wk[25 * EE * 9];           // linear: [o*576 + e*9 + tap]

#if defined(__AMDGCN__) && __has_builtin(__builtin_amdgcn_tensor_load_to_lds)
  if (threadIdx.x < 32) {
    // D# group 0: count=1, lds_addr, global_addr, type=2 ("image")
    const unsigned int lds_off = (unsigned int)(unsigned long long)&wk[0];
    const unsigned long long ga = (unsigned long long)k2w;
    u32x4 g0;
    g0.x = 1u;                                          // count=1, no gather
    g0.y = lds_off;                                     // bits[63:32]
    g0.z = (unsigned int)(ga & 0xffffffffu);            // global_addr[31:0]
    g0.w = (unsigned int)((ga >> 32) & 0x01ffffffu)     // global_addr[56:32]
           | (2u << 30);                                // type=2
    // D# group 1: data_size=4B, 1 row of 14400 elements
    i32x8 g1;
    g1[0] = (int)(2u << 16);                            // data_size=2 (4B)
    g1[1] = (int)(14400u << 16);                        // tensor_dim0[15:0]
    g1[2] = (int)(1u << 16);                            // tdim0 hi / tdim1 lo
    g1[3] = (int)(14400u << 16);                        // tdim1 hi / tile_dim0
    g1[4] = 1;                                          // tile_dim1=1
    g1[5] = 14400;                                      // tdim0_stride lo32
    g1[6] = 0;
    g1[7] = 0;
    i32x4 g2 = {};
    i32x4 g3 = {};
#if __clang_major__ >= 23
    i32x8 g4 = {};
    __builtin_amdgcn_tensor_load_to_lds(g0, g1, g2, g3, g4, 0);
#else
    __builtin_amdgcn_tensor_load_to_lds(g0, g1, g2, g3, 0);
#endif
    __builtin_amdgcn_s_wait_tensorcnt(0);
  }
#else
  for (int idx = threadIdx.x; idx < 25 * EE * 9; idx += 256) wk[idx] = k2w[idx];
#endif
  __syncthreads();

  const int tid = blockIdx.x * 256 + threadIdx.x;
  const int b   = tid / NPIX_LO;
  const int pix = tid % NPIX_LO;
  const int h = pix / WW, w = pix % WW;

  const float* cdb = cd + (size_t)b * EE * NPIX_LO;
  const float* ceb = ce + (size_t)b * EE * NPIX_HI;

  float dacc[25];
  #pragma unroll
  for (int o = 0; o < 25; ++o) dacc[o] = k2b[o];

  for (int e = 0; e < EE; ++e) {              // de branch: stride-1, pad-1 conv
    const float* cde = cdb + (size_t)e * NPIX_LO;
    #pragma unroll
    for (int tap = 0; tap < 9; ++tap) {
      const int y = h + tap / 3 - 1, x = w + tap % 3 - 1;
      const float xv = (y >= 0 && y < HH && x >= 0 && x < WW)
                           ? cde[y * WW + x] : 0.0f;
      const int widx = e * 9 + tap;
      #pragma unroll
      for (int o = 0; o < 25; ++o) dacc[o] += wk[o * 576 + widx] * xv;
    }
  }

  for (int s = 0; s < 4; ++s) {               // en branch per sub-pixel
    float eacc[25];
    #pragma unroll
    for (int o = 0; o < 25; ++o) eacc[o] = 0.0f;
    for (int e = 0; e < EE; ++e) {
      const int p = e >> 4;                   // semi-shift pad index
      const int top = 1 - (p >> 1), left = 1 - (p & 1);
      const float* cec = ceb + (size_t)(4 * (e & 15) + s) * NPIX_HI;
      #pragma unroll
      for (int tap = 0; tap < 9; ++tap) {
        const int y = 2 * h + tap / 3 - top, x = 2 * w + tap % 3 - left;
        const float xv = (y >= 0 && y < H2 && x >= 0 && x < W2)
                             ? cec[y * W2 + x] : 0.0f;
        const int widx = e * 9 + tap;
        #pragma unroll
        for (int o = 0; o < 25; ++o) eacc[o] += wk[o * 576 + widx] * xv;
      }
    }
    float mx = -1e30f;
    #pragma unroll
    for (int o = 0; o < 25; ++o) { eacc[o] += dacc[o]; mx = fmaxf(mx, eacc[o]); }
    float sum = 0.0f;
    #pragma unroll
    for (int o = 0; o < 25; ++o) { eacc[o] = __expf(eacc[o] - mx); sum += eacc[o]; }
    const float inv = 1.0f / sum;
    const int y2 = 2 * h + (s >> 1), x2 = 2 * w + (s & 1);
    float* kp = kern + (size_t)b * 25 * NPIX_HI + (size_t)y2 * W2 + x2;
    #pragma unroll
    for (int o = 0; o < 25; ++o) kp[(size_t)o * NPIX_HI] = eacc[o] * inv;
  }
}

// ---------------------------------------------------------------------------
// Kernel 4: CARAFE + gate blend. Block = 8x8 low-res tile x 4 channel lanes.
// 100 softmax weights per pixel live in registers (reused for all 256
// channels); the 12x12 de halo per channel is staged in LDS.
// ---------------------------------------------------------------------------
__global__ __launch_bounds__(256) void carafe_blend(
    const float* __restrict__ de, const float* __restrict__ en,
    const float* __restrict__ kern, const float* __restrict__ gate,
    float* __restrict__ out)
{
  __shared__ float de_lds[4][148];            // padded to dodge bank conflicts
  const int b   = blockIdx.x / 100;
  const int rem = blockIdx.x % 100;
  const int h0 = (rem / 10) * 8, w0 = (rem % 10) * 8;
  const int p  = threadIdx.x & 63;
  const int cl = threadIdx.x >> 6;
  const int py = p >> 3, px = p & 7;
  const int h = h0 + py, w = w0 + px;

  const float gv = gate[(size_t)b * NPIX_LO + h * WW + w];

  float kreg[100];
  {
    const float* kb = kern + (size_t)b * 25 * NPIX_HI;
    #pragma unroll
    for (int o = 0; o < 25; ++o)
      #pragma unroll
      for (int rs = 0; rs < 4; ++rs) {
        const int y2 = 2 * h + (rs >> 1), x2 = 2 * w + (rs & 1);
        kreg[o * 4 + rs] = kb[(size_t)o * NPIX_HI + y2 * W2 + x2];
      }
  }

  for (int cbase = 0; cbase < CC; cbase += 4) {
    const int c = cbase + cl;
    const float* dc = de + ((size_t)b * CC + c) * NPIX_LO;
    for (int idx = p; idx < 144; idx += 64) {
      const int sy = idx / 12, sx = idx % 12;
      const int gy = h0 + sy - 2, gx = w0 + sx - 2;
      de_lds[cl][idx] = (gy >= 0 && gy < HH && gx >= 0 && gx < WW)
                            ? dc[gy * WW + gx] : 0.0f;
    }
    __syncthreads();

    float acc[4] = {0.0f, 0.0f, 0.0f, 0.0f};
    #pragma unroll
    for (int i = 0; i < 5; ++i)
      #pragma unroll
      for (int j = 0; j < 5; ++j) {
        const float dv = de_lds[cl][(py + i) * 12 + (px + j)];
        const int o = i * 5 + j;
        #pragma unroll
        for (int rs = 0; rs < 4; ++rs) acc[rs] += dv * kreg[o * 4 + rs];
      }

    const float* enb = en + ((size_t)b * CC + c) * NPIX_HI;
    float* ob = out + ((size_t)b * CC + c) * NPIX_HI;
    #pragma unroll
    for (int rs = 0; rs < 4; ++rs) {
      const int y2 = 2 * h + (rs >> 1), x2 = 2 * w + (rs & 1);
      const size_t oi = (size_t)y2 * W2 + x2;
      ob[oi] = gv * enb[oi] + (1.0f - gv) * acc[rs];
    }
    __syncthreads();
  }
}

// ---------------------------------------------------------------------------
extern "C" void kernel_launch(void* const* d_in, const int* in_sizes, int n_in,
                              void* d_out, int out_size, void* d_ws, size_t ws_size,
                              hipStream_t stream) {
  (void)in_sizes; (void)n_in; (void)out_size; (void)ws_size;
  const float* en      = (const float*)d_in[0];
  const float* de      = (const float*)d_in[1];
  const float* gate_w  = (const float*)d_in[2];
  const float* gate_b  = (const float*)d_in[3];
  const float* w1_en   = (const float*)d_in[4];
  const float* b1_en   = (const float*)d_in[5];
  const float* w1_de   = (const float*)d_in[6];
  const float* conv2_k = (const float*)d_in[7];
  const float* conv2_b = (const float*)d_in[8];
  float* out = (float*)d_out;

  float* ws   = (float*)d_ws;                      // all L2-resident (192 MB)
  float* ce   = ws;                                // B*E*160*160 = 3,276,800 f
  float* cd   = ce   + (size_t)BB * EE * NPIX_HI;  // B*E*80*80   =   819,200 f
  float* gate = cd   + (size_t)BB * EE * NPIX_LO;  // B*80*80     =    12,800 f
  float* kern = gate + (size_t)BB * NPIX_LO;       // B*25*160*160= 1,280,000 f

  // ce = w1_en x en + b1_en : 4 Mtiles x 3200 Ntiles = 12800 waves
  conv1x1_wmma<NPIX_HI, true><<<3200, 128, 0, stream>>>(w1_en, en, ce, b1_en);
  // cd = w1_de x de : 4 Mtiles x 800 Ntiles = 3200 waves
  conv1x1_wmma<NPIX_LO, false><<<800, 128, 0, stream>>>(w1_de, de, cd, nullptr);
  gate_kernel<<<(BB * NPIX_LO + 255) / 256, 256, 0, stream>>>(de, gate_w,
                                                              gate_b, gate);
  semishift_kernel<<<(BB * NPIX_LO) / 256, 256, 0, stream>>>(ce, cd, conv2_k,
                                                             conv2_b, kern);
  carafe_blend<<<BB * 100, 256, 0, stream>>>(de, en, kern, gate, out);
}